// ProductQuantizer_6133213299069
// MI455X (gfx1250) — compile-verified
//
#include <hip/hip_runtime.h>
#include <stdint.h>

typedef float v2f __attribute__((ext_vector_type(2)));
typedef float v4f __attribute__((ext_vector_type(4)));
typedef float v8f __attribute__((ext_vector_type(8)));

#define B_    32
#define T_    1024
#define D_    512
#define G_    8
#define K_    256
#define GD_   64
#define NTOK  (B_ * T_)        // 32768 tokens
#define CB_STRIDE 68           // padded LDS row stride (floats): bank-conflict-free
#define TOK_PER_BLOCK 256      // 2 sub-tiles of 16 tokens per wave, 8 waves
#define NSUB  2

// ---------------------------------------------------------------------------
// Kernel 1: per-code squared norms -> ws, and zero the loss scalar in d_out.
// ---------------------------------------------------------------------------
__global__ void pq_csq_init(const float* __restrict__ cb,
                            float* __restrict__ csq,
                            float* __restrict__ loss) {
  const int g = blockIdx.x;        // 8 blocks
  const int k = threadIdx.x;       // 256 threads
  const float* row = cb + ((size_t)g * K_ + k) * GD_;
  float s = 0.f;
  #pragma unroll
  for (int i = 0; i < GD_; i += 4) {
    v4f v = *(const v4f*)(row + i);
    s += v.x * v.x + v.y * v.y + v.z * v.z + v.w * v.w;
  }
  csq[g * K_ + k] = s;
  if (g == 0 && k == 0) *loss = 0.f;
}

// ---------------------------------------------------------------------------
// Kernel 2: main PQ kernel. Grid: (NTOK/256, G). Block: 256 threads = 8 waves.
// Codebook staged via GLOBAL_LOAD_ASYNC_TO_LDS_B128 (ASYNCcnt path); each wave
// runs 2 sub-tiles of 16 tokens x 256 codes via V_WMMA_F32_16X16X4_F32.
// ---------------------------------------------------------------------------
__global__ void __launch_bounds__(256)
pq_main(const float* __restrict__ x,
        const float* __restrict__ cb,
        const float* __restrict__ csq_g,
        float* __restrict__ out_xq,
        float* __restrict__ loss,
        int* __restrict__ out_idx) {
  // 256 codes x 68-float padded rows = 69,632 B (LDS is 320 KB/WGP on CDNA5)
  __shared__ float lds_cb[K_ * CB_STRIDE];

  const int g   = blockIdx.y;
  const int tid = threadIdx.x;

  // ---- async-stage this group's codebook into LDS (memory -> LDS direct) ----
  // 16384 floats = 4096 x b128; 256 threads x 16 chunks, padded row layout.
  const float* cbg = cb + (size_t)g * K_ * GD_;
  #pragma unroll
  for (int i = 0; i < 16; ++i) {
    int e4   = i * 256 + tid;      // float4 chunk index in [0, 4096)
    int code = e4 >> 4;            // 16 chunks per 64-float code row
    int col4 = e4 & 15;
    // low 32 bits of a generic pointer to LDS = byte offset from LDS base
    unsigned lds_off = (unsigned)(uintptr_t)&lds_cb[code * CB_STRIDE + col4 * 4];
    unsigned goff    = (unsigned)((code * GD_ + col4 * 4) * sizeof(float));
    asm volatile("global_load_async_to_lds_b128 %0, %1, %2"
                 :: "v"(lds_off), "v"(goff), "s"(cbg) : "memory");
  }
  asm volatile("s_wait_asynccnt 0x0" ::: "memory");
  __syncthreads();

  const int wave = tid >> 5;
  const int lane = tid & 31;
  const int l16  = lane & 15;
  const int h    = lane >> 4;      // half of the wave (K/M split in WMMA layout)

  const float* csq = csq_g + g * K_;
  float lloss = 0.f;

  for (int sub = 0; sub < NSUB; ++sub) {
    const int tokenBase = blockIdx.x * TOK_PER_BLOCK + sub * 128 + wave * 16;
    const int tokA      = tokenBase + l16;          // A-matrix row (token)

    // ---- load A: 16 tokens x 64 dims in the 16x4 f32 WMMA lane layout ----
    // lane<16 holds dims {4s,4s+1}; lane>=16 holds dims {4s+2,4s+3}
    const float* xrow = x + (size_t)tokA * D_ + g * GD_ + 2 * h;
    v2f a[16];
    float xsq = 0.f;
    #pragma unroll
    for (int s = 0; s < 16; ++s) {
      a[s] = *(const v2f*)(xrow + 4 * s);
      xsq += a[s].x * a[s].x + a[s].y * a[s].y;
    }
    xsq += __shfl_xor(xsq, 16, 32);  // full ||x_g||^2 of token tokA, both halves

    float bestVal[8];                // running (csq - 2*dot) per lane column
    int   bestIdx[8];
    #pragma unroll
    for (int r = 0; r < 8; ++r) { bestVal[r] = 3.4e38f; bestIdx[r] = 0; }

    // ---- 16 N-tiles of 16 codes; each = 16 chained f32 WMMAs over K=64 ----
    for (int t = 0; t < 16; ++t) {
      const int n = t * 16 + l16;                  // code column of this lane
      const float* brow = &lds_cb[n * CB_STRIDE + 2 * h];
      v8f c = {0.f, 0.f, 0.f, 0.f, 0.f, 0.f, 0.f, 0.f};
      #pragma unroll
      for (int s = 0; s < 16; ++s) {
        v2f bv = *(const v2f*)(brow + 4 * s);      // ds_load_2addr_b64 pairs
        c = __builtin_amdgcn_wmma_f32_16x16x4_f32(false, a[s], false, bv,
                                                  (short)0, c, false, false);
      }
      const float cs = csq[n];
      #pragma unroll
      for (int r = 0; r < 8; ++r) {                // rows m = r + 8*h
        float val = cs - 2.f * c[r];               // xsq is constant in n
        if (val < bestVal[r]) { bestVal[r] = val; bestIdx[r] = n; }
      }
    }

    // ---- argmin across the 16 lanes of each half (tie -> lower index) ----
    #pragma unroll
    for (int mask = 1; mask <= 8; mask <<= 1) {
      #pragma unroll
      for (int r = 0; r < 8; ++r) {
        float ov = __shfl_xor(bestVal[r], mask, 32);
        int   oi = __shfl_xor(bestIdx[r], mask, 32);
        if (ov < bestVal[r] || (ov == bestVal[r] && oi < bestIdx[r])) {
          bestVal[r] = ov; bestIdx[r] = oi;
        }
      }
    }
    // every lane of half h now holds bestVal/bestIdx for rows m = r + 8*h

    // ---- loss: minDist = ||x||^2 + best(csq - 2*dot); xsq[m] at lane m ----
    #pragma unroll
    for (int r = 0; r < 8; ++r) {
      float xs = __shfl(xsq, r + 8 * h, 32);
      lloss += xs + bestVal[r];
    }

    // ---- indices ----
    if (l16 == 0) {                                // one lane per half
      #pragma unroll
      for (int r = 0; r < 8; ++r) {
        int tok = tokenBase + 8 * h + r;
        out_idx[tok * G_ + g] = bestIdx[r];
      }
    }

    // ---- x_q = selected code (straight-through forward). Half h writes its
    // 8 tokens; lane l16 covers 4 contiguous floats of the 64-float row. ----
    #pragma unroll
    for (int r = 0; r < 8; ++r) {
      int tok = tokenBase + 8 * h + r;
      v4f q = *(const v4f*)(&lds_cb[bestIdx[r] * CB_STRIDE + 4 * l16]);
      *(v4f*)(out_xq + (size_t)tok * D_ + g * GD_ + 4 * l16) = q;
    }
  }

  // one pre-scaled loss atomic per half-wave (lanes 0 and 16)
  if (l16 == 0) {
    atomicAdd(loss, lloss * (2.0f / ((float)NTOK * (float)GD_)));
  }
}

// ---------------------------------------------------------------------------
extern "C" void kernel_launch(void* const* d_in, const int* in_sizes, int n_in,
                              void* d_out, int out_size, void* d_ws, size_t ws_size,
                              hipStream_t stream) {
  const float* x  = (const float*)d_in[0];   // [B,T,D] f32
  const float* cb = (const float*)d_in[1];   // [G,K,GD] f32

  float* out_xq  = (float*)d_out;                       // [B,T,D]
  float* loss    = out_xq + (size_t)NTOK * D_;          // scalar
  int*   out_idx = (int*)(loss + 1);                    // [B,T,G] int32

  float* csq = (float*)d_ws;                            // [G,K] scratch

  pq_csq_init<<<dim3(G_), dim3(K_), 0, stream>>>(cb, csq, loss);
  pq_main<<<dim3(NTOK / TOK_PER_BLOCK, G_), dim3(256), 0, stream>>>(
      x, cb, csq, out_xq, loss, out_idx);
}